// ModelwP_57698590654560
// MI455X (gfx1250) — compile-verified
//
#include <hip/hip_runtime.h>
#include <cmath>
#include <stdint.h>

#define EPS 1e-5f

static constexpr int Nn = 32, T0 = 64, Vj = 25, Mp = 2, C0 = 3, Hh = 8;
static constexpr int NM = Nn * Mp;           // 64
static constexpr int NCLASS = 60;
static constexpr int FCC = 256;

typedef float v2f __attribute__((ext_vector_type(2)));
typedef float v8f __attribute__((ext_vector_type(8)));

struct BN4 { const float* beta; const float* gamma; const float* mean; const float* var; };

__device__ __forceinline__ float bn_apply(float x, const BN4& b, int c) {
  float g = b.gamma[c] * rsqrtf(b.var[c] + EPS);
  return (x - b.mean[c]) * g + b.beta[c];
}

// ---------------------------------------------------------------------------
// Input transform: xv = A_vector @ x  (per frame, per person), then data_bn,
// producing h0 laid out (NM, C0, T0, Vj).
// ---------------------------------------------------------------------------
__global__ void k_input(const float* __restrict__ x, const float* __restrict__ Avec,
                        BN4 dbn, float* __restrict__ h)
{
  int idx = blockIdx.x * blockDim.x + threadIdx.x;
  if (idx >= NM * T0 * Vj) return;
  int v = idx % Vj;
  int t = (idx / Vj) % T0;
  int nm = idx / (Vj * T0);
  int n = nm / Mp, m = nm % Mp;
  for (int c = 0; c < C0; ++c) {
    float s = 0.f;
    for (int vp = 0; vp < Vj; ++vp)
      s += Avec[v * Vj + vp] * x[(((n * C0 + c) * T0 + t) * Vj + vp) * Mp + m];
    int ch = (m * Vj + v) * C0 + c;
    h[((nm * C0 + c) * T0 + t) * Vj + v] = bn_apply(s, dbn, ch);
  }
}

// ---------------------------------------------------------------------------
// AGCN: per frame b=(n',t), per subset s: Z = (PA_s * attn_b_s) @ X, then
// Y += Z @ Wd_sT.  WMMA f32 16x16x4 throughout, fused BN + residual + GELU.
// Block = 8 waves (256 threads), one block per frame.
// X tile staged to LDS via the CDNA5 async-to-LDS engine (ASYNCcnt); the
// next subset's attention tile is prefetched (global_prefetch_b8) so the
// dominant attention stream overlaps the WMMA work.
// ---------------------------------------------------------------------------
#define XS 260
#define ZS 132
#define AS 33

__global__ __launch_bounds__(256)
void k_agcn(const float* __restrict__ hin, const float* __restrict__ attn,
            const float* __restrict__ PA, const float* __restrict__ Wd,
            const float* __restrict__ bd, BN4 bn,
            const float* __restrict__ down_w, const float* __restrict__ down_b,
            BN4 dbn, int hasDown,
            float* __restrict__ gout, int C, int O, int T, int dsLvl)
{
  __shared__ float Xs[32 * XS];   // X[v][c], v padded to 32, zero-filled
  __shared__ float Zs[32 * ZS];   // Z[u][c] chunk (<=128 cols)
  __shared__ float As[32 * AS];   // A_s[u][v]

  int b = blockIdx.x;
  int t = b % T, np = b / T;
  int tid = threadIdx.x;
  int lane = tid & 31, wave = tid >> 5;
  int Cp = (C + 15) & ~15;

  for (int i = tid; i < 32 * XS; i += 256) Xs[i] = 0.f;
  __syncthreads();
  // async DMA: global -> LDS for the X tile (tracked by ASYNCcnt)
  for (int i = tid; i < Vj * C; i += 256) {
    int v = i / C, c = i % C;
    const float* gp = hin + ((size_t)(np * C + c) * T + t) * Vj + v;
    unsigned lo = (unsigned)(uintptr_t)(&Xs[v * XS + c]);
    asm volatile("global_load_async_to_lds_b32 %0, %1, off"
                 :: "v"(lo), "v"(gp) : "memory");
  }
  asm volatile("s_wait_asynccnt 0" ::: "memory");

  int totTiles = 2 * (O >> 4);   // 8 / 16 / 32
  v8f yacc[4];
  for (int q = 0; q < 4; ++q)
    yacc[q] = (v8f){0.f, 0.f, 0.f, 0.f, 0.f, 0.f, 0.f, 0.f};

  int reps = 1 << dsLvl, stepT = 64 >> dsLvl;
  float wrep = 1.0f / (float)reps;

  for (int s = 0; s < Hh; ++s) {
    // build A_s = PA_s * mean-of-replicas(attn)
    for (int i = tid; i < 32 * 32; i += 256) {
      int u = i >> 5, v = i & 31;
      float a = 0.f;
      if (u < Vj && v < Vj) {
        float sum = 0.f;
        for (int j = 0; j < reps; ++j)
          sum += attn[(size_t)(np * 64 + t + j * stepT) * (Hh * 625) + s * 625 + u * 25 + v];
        a = PA[s * 625 + u * 25 + v] * sum * wrep;
      }
      As[u * AS + v] = a;
    }
    // warm L2/L0 for the next subset's attention tile (global_prefetch_b8)
    if (s + 1 < Hh) {
      int nline = (reps * 625 + 31) >> 5;
      for (int i = tid; i < nline; i += 256) {
        int e = i << 5;
        int j = e / 625, r = e % 625;
        __builtin_prefetch(&attn[(size_t)(np * 64 + t + j * stepT) * (Hh * 625) +
                                 (s + 1) * 625 + r], 0, 1);
      }
    }
    __syncthreads();

    for (int cc = 0; cc < Cp; cc += 128) {
      int chunk = (Cp - cc < 128) ? (Cp - cc) : 128;
      // ----- Z = A_s @ X[:, cc:cc+chunk] -----
      int zt = 2 * (chunk >> 4);
      for (int ti = wave; ti < zt; ti += 8) {
        int mu = (ti & 1) * 16, nb = (ti >> 1) * 16;
        v8f zacc = (v8f){0.f, 0.f, 0.f, 0.f, 0.f, 0.f, 0.f, 0.f};
        for (int k = 0; k < 28; k += 4) {
          int kk = k + (lane >> 4) * 2;
          v2f af, bf;
          af.x = As[(mu + (lane & 15)) * AS + kk];
          af.y = As[(mu + (lane & 15)) * AS + kk + 1];
          bf.x = Xs[kk * XS + cc + nb + (lane & 15)];
          bf.y = Xs[(kk + 1) * XS + cc + nb + (lane & 15)];
          zacc = __builtin_amdgcn_wmma_f32_16x16x4_f32(false, af, false, bf,
                                                       (short)0, zacc, false, false);
        }
        for (int r = 0; r < 8; ++r)
          Zs[(mu + r + 8 * (lane >> 4)) * ZS + nb + (lane & 15)] = zacc[r];
      }
      __syncthreads();
      // ----- Y += Z @ Wd_s^T -----
      for (int q = 0; q < 4; ++q) {
        int ti = wave + q * 8;
        if (ti < totTiles) {
          int mu = (ti & 1) * 16, ob = (ti >> 1) * 16;
          int o = ob + (lane & 15);
          v8f acc = yacc[q];
          for (int k = 0; k < chunk; k += 4) {
            int kk = k + (lane >> 4) * 2;
            v2f af, bf;
            af.x = Zs[(mu + (lane & 15)) * ZS + kk];
            af.y = Zs[(mu + (lane & 15)) * ZS + kk + 1];
            int c1 = cc + kk, c2 = cc + kk + 1;
            bf.x = (c1 < C) ? Wd[(size_t)(s * O + o) * C + c1] : 0.f;
            bf.y = (c2 < C) ? Wd[(size_t)(s * O + o) * C + c2] : 0.f;
            acc = __builtin_amdgcn_wmma_f32_16x16x4_f32(false, af, false, bf,
                                                        (short)0, acc, false, false);
          }
          yacc[q] = acc;
        }
      }
      __syncthreads();
    }
  }

  // epilogue: bias sum, BN, residual (identity from Xs or down-proj), GELU
  for (int q = 0; q < 4; ++q) {
    int ti = wave + q * 8;
    if (ti >= totTiles) continue;
    int mu = (ti & 1) * 16, ob = (ti >> 1) * 16;
    int o = ob + (lane & 15);
    float bsum = 0.f;
    for (int s2 = 0; s2 < Hh; ++s2) bsum += bd[s2 * O + o];
    float bg = bn.gamma[o] * rsqrtf(bn.var[o] + EPS);
    for (int r = 0; r < 8; ++r) {
      int u = mu + r + 8 * (lane >> 4);
      if (u >= Vj) continue;
      float y = yacc[q][r] + bsum;
      y = (y - bn.mean[o]) * bg + bn.beta[o];
      float res;
      if (hasDown) {
        float rs = down_b[o];
        for (int c = 0; c < C; ++c) rs += down_w[o * C + c] * Xs[u * XS + c];
        res = bn_apply(rs, dbn, o);
      } else {
        res = Xs[u * XS + o];      // cin == cout: identity residual
      }
      float z = y + res;
      float gel = 0.5f * z * (1.0f + erff(z * 0.70710678118f));
      gout[((size_t)(np * O + o) * T + t) * Vj + u] = gel;
    }
  }
}

// ---------------------------------------------------------------------------
// Generic WMMA 1x1 conv: out[p, oc] = in[:, p]^T @ W^T, fused bias+BN(+ReLU).
// One wave per 16x16 output tile.  Optional temporal stride on the input.
// ---------------------------------------------------------------------------
__global__ __launch_bounds__(32)
void k_conv1x1(const float* __restrict__ in, const float* __restrict__ W,
               const float* __restrict__ bias, BN4 bn, float* __restrict__ out,
               int Cin, int Tin, int OC, int ochStride, int ochOff,
               int Tout, int tstride, int doRelu)
{
  int lane = threadIdx.x;
  int TVo = Tout * Vj;
  int PT = TVo >> 4, OT = OC >> 4;
  int bid = blockIdx.x;
  int pt = bid % PT; bid /= PT;
  int ot = bid % OT;
  int np = bid / OT;
  int pbase = pt * 16, obase = ot * 16;

  int rowp = pbase + (lane & 15);
  int tA = (rowp / Vj) * tstride, vA = rowp % Vj;
  size_t cstride = (size_t)Tin * Vj;
  const float* inA = in + (size_t)np * Cin * cstride + (size_t)tA * Vj + vA;
  const float* Wb = W + (size_t)(obase + (lane & 15)) * Cin;
  int khalf = (lane >> 4) * 2;

  v8f acc = (v8f){0.f, 0.f, 0.f, 0.f, 0.f, 0.f, 0.f, 0.f};
  for (int k = 0; k < Cin; k += 4) {
    int kk = k + khalf;
    v2f af, bf;
    af.x = inA[(size_t)kk * cstride];
    af.y = inA[(size_t)(kk + 1) * cstride];
    bf.x = Wb[kk];
    bf.y = Wb[kk + 1];
    acc = __builtin_amdgcn_wmma_f32_16x16x4_f32(false, af, false, bf,
                                                (short)0, acc, false, false);
  }

  int oc = obase + (lane & 15);
  float bg = bn.gamma[oc] * rsqrtf(bn.var[oc] + EPS);
  float bb = bias[oc];
  for (int r = 0; r < 8; ++r) {
    int p = pbase + r + 8 * (lane >> 4);
    int t = p / Vj, v = p % Vj;
    float y = acc[r] + bb;
    y = (y - bn.mean[oc]) * bg + bn.beta[oc];
    if (doRelu) y = fmaxf(y, 0.f);
    out[(((size_t)np * ochStride + ochOff + oc) * Tout + t) * Vj + v] = y;
  }
}

// ---------------------------------------------------------------------------
// Temporal conv k=5 (full channel mix), 'same' pad, stride/dilation, + BN.
// ---------------------------------------------------------------------------
__global__ void k_tconv(const float* __restrict__ pre, const float* __restrict__ wt,
                        const float* __restrict__ bt, BN4 bn, float* __restrict__ out,
                        int bc, int Tin, int Tout, int stride, int dil,
                        int ochStride, int ochOff)
{
  int idx = blockIdx.x * blockDim.x + threadIdx.x;
  int total = NM * bc * Tout * Vj;
  if (idx >= total) return;
  int v = idx % Vj;
  int t = (idx / Vj) % Tout;
  int oc = (idx / (Vj * Tout)) % bc;
  int np = idx / (Vj * Tout * bc);
  int pad = 2 * dil;
  float acc = bt[oc];
  for (int kt = 0; kt < 5; ++kt) {
    int tin = t * stride - pad + kt * dil;
    if (tin < 0 || tin >= Tin) continue;
    const float* prow = pre + ((size_t)np * bc * Tin + tin) * Vj + v;
    const float* wrow = wt + (size_t)(oc * bc) * 5 + kt;
    for (int ic = 0; ic < bc; ++ic)
      acc += wrow[(size_t)ic * 5] * prow[(size_t)ic * Tin * Vj];
  }
  float y = bn_apply(acc, bn, oc);
  out[(((size_t)np * ochStride + ochOff + oc) * Tout + t) * Vj + v] = y;
}

// ---------------------------------------------------------------------------
// MaxPool (3,1) stride (s,1) pad (1,0), + BN.
// ---------------------------------------------------------------------------
__global__ void k_maxpool(const float* __restrict__ pre, BN4 bn, float* __restrict__ out,
                          int bc, int Tin, int Tout, int stride, int ochStride, int ochOff)
{
  int idx = blockIdx.x * blockDim.x + threadIdx.x;
  int total = NM * bc * Tout * Vj;
  if (idx >= total) return;
  int v = idx % Vj;
  int t = (idx / Vj) % Tout;
  int oc = (idx / (Vj * Tout)) % bc;
  int np = idx / (Vj * Tout * bc);
  float m = -3.402823e38f;
  for (int dt = -1; dt <= 1; ++dt) {
    int tin = t * stride + dt;
    if (tin >= 0 && tin < Tin)
      m = fmaxf(m, pre[(((size_t)np * bc + oc) * Tin + tin) * Vj + v]);
  }
  float y = bn_apply(m, bn, oc);
  out[(((size_t)np * ochStride + ochOff + oc) * Tout + t) * Vj + v] = y;
}

// ---------------------------------------------------------------------------
// Layer residual + ReLU (in-place on the concatenated TCN output).
// kind: 0=none, 1=identity, 2=strided 1x1 conv + BN.
// ---------------------------------------------------------------------------
__global__ void k_res_relu(float* __restrict__ hout, const float* __restrict__ hin,
                           const float* __restrict__ res_w, const float* __restrict__ res_b,
                           BN4 rbn, int kind, int Cin, int O, int Tin, int Tout, int stride)
{
  int idx = blockIdx.x * blockDim.x + threadIdx.x;
  int total = NM * O * Tout * Vj;
  if (idx >= total) return;
  int v = idx % Vj;
  int t = (idx / Vj) % Tout;
  int o = (idx / (Vj * Tout)) % O;
  int np = idx / (Vj * Tout * O);
  size_t pos = (((size_t)np * O + o) * Tout + t) * Vj + v;
  float val = hout[pos];
  float r = 0.f;
  if (kind == 1) {
    r = hin[(((size_t)np * O + o) * Tin + t) * Vj + v];
  } else if (kind == 2) {
    float rs = res_b[o];
    size_t basei = (size_t)np * Cin * Tin * Vj + (size_t)(t * stride) * Vj + v;
    for (int c = 0; c < Cin; ++c)
      rs += res_w[(size_t)o * Cin + c] * hin[basei + (size_t)c * Tin * Vj];
    r = bn_apply(rs, rbn, o);
  }
  hout[pos] = fmaxf(val + r, 0.f);
}

// ---------------------------------------------------------------------------
// Global average pool over (M, T, V) and the final FC.
// ---------------------------------------------------------------------------
__global__ void k_pool(const float* __restrict__ h, float* __restrict__ pooled, int Tf)
{
  int idx = blockIdx.x * blockDim.x + threadIdx.x;
  if (idx >= Nn * FCC) return;
  int ch = idx % FCC, n = idx / FCC;
  float s = 0.f;
  for (int m = 0; m < Mp; ++m)
    for (int t = 0; t < Tf; ++t)
      for (int v = 0; v < Vj; ++v)
        s += h[(((size_t)(n * Mp + m) * FCC + ch) * Tf + t) * Vj + v];
  pooled[idx] = s / (float)(Mp * Tf * Vj);
}

__global__ void k_fc(const float* __restrict__ pooled, const float* __restrict__ fw,
                     const float* __restrict__ fb, float* __restrict__ out)
{
  int idx = blockIdx.x * blockDim.x + threadIdx.x;
  if (idx >= Nn * NCLASS) return;
  int c = idx % NCLASS, n = idx / NCLASS;
  float s = fb[c];
  for (int k = 0; k < FCC; ++k) s += pooled[n * FCC + k] * fw[c * FCC + k];
  out[idx] = s;
}

// ---------------------------------------------------------------------------
// Host orchestration
// ---------------------------------------------------------------------------
struct LayerP {
  const float *PA, *Wd, *bd; BN4 bn;
  const float *down_b, *down_w; BN4 dbn; int hasDown;
  const float *res_b, *res_w; BN4 rbn;
  const float *b1_b, *b1_bt, *b1_w, *b1_wt; BN4 b1_bn1, b1_bn2;
  const float *b2_b, *b2_bt, *b2_w, *b2_wt; BN4 b2_bn1, b2_bn2;
  const float *b3_b, *b3_w; BN4 b3_bn1, b3_bn2;
  const float *b4_b, *b4_w; BN4 b4_bn1;
};

extern "C" void kernel_launch(void* const* d_in, const int* in_sizes, int n_in,
                              void* d_out, int out_size, void* d_ws, size_t ws_size,
                              hipStream_t stream)
{
  (void)in_sizes; (void)n_in; (void)out_size; (void)ws_size;
  int ii = 0;
  auto P = [&]() { return (const float*)d_in[ii++]; };

  const float* x    = P();
  const float* attn = P();
  const float* Avec = P();
  // params, jax tree_flatten order (sorted dict keys)
  BN4 data_bn{P(), P(), P(), P()};      // beta, gamma, mean, var
  const float* fc_b = P();
  const float* fc_w = P();

  const int cin_[10]  = {3, 64, 64, 64, 64, 128, 128, 128, 256, 256};
  const int cout_[10] = {64, 64, 64, 64, 128, 128, 128, 256, 256, 256};
  const int str_[10]  = {1, 1, 1, 1, 2, 1, 1, 2, 1, 1};
  const int res_[10]  = {0, 1, 1, 1, 2, 1, 1, 2, 1, 1};
  const int Tin_[10]  = {64, 64, 64, 64, 64, 32, 32, 32, 16, 16};
  const int ds_[10]   = {0, 0, 0, 0, 0, 1, 1, 1, 2, 2};

  LayerP Lp[10] = {};
  for (int l = 0; l < 10; ++l) {
    LayerP& q = Lp[l];
    q.PA = P(); q.Wd = P(); q.bd = P(); q.bn = {P(), P(), P(), P()};
    q.hasDown = (cin_[l] != cout_[l]);
    if (q.hasDown) { q.down_b = P(); q.dbn = {P(), P(), P(), P()}; q.down_w = P(); }
    if (res_[l] == 2) { q.res_b = P(); q.rbn = {P(), P(), P(), P()}; q.res_w = P(); }
    q.b1_b = P(); q.b1_bn1 = {P(), P(), P(), P()}; q.b1_bn2 = {P(), P(), P(), P()};
    q.b1_bt = P(); q.b1_w = P(); q.b1_wt = P();
    q.b2_b = P(); q.b2_bn1 = {P(), P(), P(), P()}; q.b2_bn2 = {P(), P(), P(), P()};
    q.b2_bt = P(); q.b2_w = P(); q.b2_wt = P();
    q.b3_b = P(); q.b3_bn1 = {P(), P(), P(), P()}; q.b3_bn2 = {P(), P(), P(), P()};
    q.b3_w = P();
    q.b4_b = P(); q.b4_bn1 = {P(), P(), P(), P()}; q.b4_w = P();
  }

  float* base = (float*)d_ws;
  const size_t ACT  = (size_t)NM * 4096 * Vj;    // 6,553,600 floats (C*T <= 4096)
  const size_t PREM = (size_t)NM * 64 * 32 * Vj; // 3,276,800 floats
  float* bufA = base;
  float* bufB = base + ACT;
  float* bufG = base + 2 * ACT;
  float* preB[3] = { base + 3 * ACT, base + 3 * ACT + PREM, base + 3 * ACT + 2 * PREM };
  float* pooled = base + 3 * ACT + 3 * PREM;

  {
    int tot = NM * T0 * Vj;
    k_input<<<(tot + 255) / 256, 256, 0, stream>>>(x, Avec, data_bn, bufA);
  }

  float* hc = bufA;
  float* hn = bufB;
  for (int l = 0; l < 10; ++l) {
    int C = cin_[l], O = cout_[l], st = str_[l];
    int Tin = Tin_[l], Tout = Tin / st, bc = O / 4;
    LayerP& q = Lp[l];

    k_agcn<<<NM * Tin, 256, 0, stream>>>(hc, attn, q.PA, q.Wd, q.bd, q.bn,
                                         q.down_w, q.down_b, q.dbn, q.hasDown,
                                         bufG, C, O, Tin, ds_[l]);

    int TVi = Tin * Vj, TVo = Tout * Vj;
    dim3 g1(NM * (TVi / 16) * (bc / 16));
    k_conv1x1<<<g1, 32, 0, stream>>>(bufG, q.b1_w, q.b1_b, q.b1_bn1, preB[0],
                                     O, Tin, bc, bc, 0, Tin, 1, 1);
    k_conv1x1<<<g1, 32, 0, stream>>>(bufG, q.b2_w, q.b2_b, q.b2_bn1, preB[1],
                                     O, Tin, bc, bc, 0, Tin, 1, 1);
    k_conv1x1<<<g1, 32, 0, stream>>>(bufG, q.b3_w, q.b3_b, q.b3_bn1, preB[2],
                                     O, Tin, bc, bc, 0, Tin, 1, 1);
    dim3 g4(NM * (TVo / 16) * (bc / 16));
    k_conv1x1<<<g4, 32, 0, stream>>>(bufG, q.b4_w, q.b4_b, q.b4_bn1, hn,
                                     O, Tin, bc, O, 3 * bc, Tout, st, 0);

    int totb = NM * bc * Tout * Vj;
    k_tconv<<<(totb + 255) / 256, 256, 0, stream>>>(preB[0], q.b1_wt, q.b1_bt, q.b1_bn2,
                                                    hn, bc, Tin, Tout, st, 1, O, 0);
    k_tconv<<<(totb + 255) / 256, 256, 0, stream>>>(preB[1], q.b2_wt, q.b2_bt, q.b2_bn2,
                                                    hn, bc, Tin, Tout, st, 2, O, bc);
    k_maxpool<<<(totb + 255) / 256, 256, 0, stream>>>(preB[2], q.b3_bn2, hn,
                                                      bc, Tin, Tout, st, O, 2 * bc);

    int toto = NM * O * Tout * Vj;
    k_res_relu<<<(toto + 255) / 256, 256, 0, stream>>>(hn, hc, q.res_w, q.res_b, q.rbn,
                                                       res_[l], C, O, Tin, Tout, st);
    float* tmp = hc; hc = hn; hn = tmp;
  }

  k_pool<<<(Nn * FCC + 255) / 256, 256, 0, stream>>>(hc, pooled, 16);
  k_fc<<<(Nn * NCLASS + 255) / 256, 256, 0, stream>>>(pooled, fc_w, fc_b, (float*)d_out);
}